// EgoGNN_5720896438655
// MI455X (gfx1250) — compile-verified
//
#include <hip/hip_runtime.h>
#include <hip/hip_bf16.h>

typedef float v2f __attribute__((ext_vector_type(2)));
typedef float v8f __attribute__((ext_vector_type(8)));

// ---------------------------------------------------------------------------
// Degree / normalization:  deg[i] = 1 + #(dst==i);  dinv[i] = rsqrt(deg[i])
// ---------------------------------------------------------------------------
__global__ void deg_init_kernel(float* __restrict__ deg, int n) {
  int i = blockIdx.x * blockDim.x + threadIdx.x;
  if (i < n) deg[i] = 1.0f;                      // self-loop
}

__global__ void deg_count_kernel(const int* __restrict__ dst,
                                 float* __restrict__ deg, int E) {
  int e = blockIdx.x * blockDim.x + threadIdx.x;
  if (e < E) {
    int d = __builtin_nontemporal_load(&dst[e]); // stream edge list (NT hint)
    atomicAdd(&deg[d], 1.0f);
  }
}

__global__ void deg_finalize_kernel(float* __restrict__ deg, int n) {
  int i = blockIdx.x * blockDim.x + threadIdx.x;
  if (i < n) deg[i] = rsqrtf(deg[i]);
}

// ---------------------------------------------------------------------------
// Dense transform H[M,N] = X[M,K] @ W[K,N] via V_WMMA_F32_16X16X4_F32.
// One wave owns 16 rows x all N columns (TN=N/16 accumulator tiles): the A
// fragment is loaded once per k-step and reused across tiles. K,N are
// compile-time so the k-loop fully unrolls; the K=5 tail is handled with
// clamped indices + cndmask scales (no EXEC divergence: WMMA needs EXEC=~0).
// Requires M % 16 == 0 (holds: M = 100000).
// A layout (16x4 f32): lane l -> row l&15, K-pair 2*(l>>4)+{0,1}.
// B layout (4x16 f32): VGPR r: lanes<16 K=r, lanes>=16 K=r+2; col l&15.
// C/D (16x16 f32): VGPR r: lanes<16 M=r, lanes>=16 M=r+8; col l&15.
// ---------------------------------------------------------------------------
template <int K, int N>
__launch_bounds__(256)
__global__ void gemm_wmma_kernel(const float* __restrict__ X,
                                 const float* __restrict__ W,
                                 float* __restrict__ H, int M) {
  constexpr int TN = N / 16;
  const int lane = threadIdx.x & 31;
  const int tm = blockIdx.x * 8 + (threadIdx.x >> 5); // wave-uniform tile row
  if (tm * 16 >= M) return;                           // uniform exit

  const int nsub = lane & 15;
  const int kh = lane >> 4;                           // 0 or 1
  const float* __restrict__ Xr = X + (tm * 16 + nsub) * K;   // this lane's A row

  v8f acc[TN];
#pragma unroll
  for (int t = 0; t < TN; ++t) acc[t] = (v8f){};

#pragma unroll
  for (int k0 = 0; k0 < K; k0 += 4) {
    const int ka = k0 + 2 * kh;                       // this lane's first K
    v2f a;
    float s0 = 1.0f, s1 = 1.0f;
    int i0 = ka, i1 = ka + 1;
    if (k0 + 3 < K || (K & 3) == 0) {                 // compile-time per iter
      if ((K & 1) == 0) {                             // rows 8B-aligned
        a = *(const v2f*)(Xr + ka);
      } else {
        a.x = Xr[ka]; a.y = Xr[ka + 1];
      }
    } else {                                          // K tail (layer 1, K=5)
      s0 = (ka     < K) ? 1.0f : 0.0f;
      s1 = (ka + 1 < K) ? 1.0f : 0.0f;
      i0 = (ka     < K) ? ka     : 0;                 // clamp: always in-bounds
      i1 = (ka + 1 < K) ? ka + 1 : 0;
      a.x = Xr[i0] * s0;
      a.y = Xr[i1] * s1;
    }
#pragma unroll
    for (int t = 0; t < TN; ++t) {
      v2f b;
      b.x = W[i0 * N + t * 16 + nsub] * s0;           // const offsets from W+nsub
      b.y = W[i1 * N + t * 16 + nsub] * s1;
      acc[t] = __builtin_amdgcn_wmma_f32_16x16x4_f32(
          false, a, false, b, (short)0, acc[t], false, false);
    }
  }

  const int rowBase = tm * 16 + 8 * kh;
#pragma unroll
  for (int r = 0; r < 8; ++r) {
    float* __restrict__ Hr = H + (rowBase + r) * N + nsub;
#pragma unroll
    for (int t = 0; t < TN; ++t) Hr[t * 16] = acc[t][r];
  }
}

// ---------------------------------------------------------------------------
// agg[i,:] = dinv[i]^2 * h[i,:]   (self-loop contribution, also inits agg)
// ---------------------------------------------------------------------------
template <int F>
__global__ void self_init_kernel(const float* __restrict__ h,
                                 const float* __restrict__ dinv,
                                 float* __restrict__ agg, int n) {
  unsigned idx = blockIdx.x * blockDim.x + threadIdx.x;
  if (idx >= (unsigned)n * F) return;
  unsigned i = idx / F;                          // F power of two -> shift
  float di = dinv[i];
  agg[idx] = di * di * h[idx];
}

// ---------------------------------------------------------------------------
// Edge scatter: F/4 consecutive lanes cover one edge; float4 gather of
// h[src] is fully coalesced; 4 L2 float atomics per lane into agg[dst].
// Working set (25.6 MB) is resident in the 192 MB L2.
// ---------------------------------------------------------------------------
template <int F>
__global__ void edge_scatter_kernel(const float* __restrict__ h,
                                    float* __restrict__ agg,
                                    const int* __restrict__ src,
                                    const int* __restrict__ dst,
                                    const float* __restrict__ dinv, int E) {
  constexpr unsigned CH = F / 4;                 // float4 chunks per edge
  unsigned idx = blockIdx.x * blockDim.x + threadIdx.x;
  unsigned e = idx / CH;                         // shift
  if (e >= (unsigned)E) return;
  unsigned q = idx & (CH - 1);

  int s = __builtin_nontemporal_load(&src[e]);   // edge list streamed once
  int d = __builtin_nontemporal_load(&dst[e]);
  float w = dinv[s] * dinv[d];

  float4 v = *((const float4*)(h + (size_t)s * F) + q);  // L2-resident gather
  float* o = agg + (size_t)d * F + (size_t)q * 4;
  atomicAdd(o + 0, w * v.x);
  atomicAdd(o + 1, w * v.y);
  atomicAdd(o + 2, w * v.z);
  atomicAdd(o + 3, w * v.w);
}

// ---------------------------------------------------------------------------
// out[i,f] = relu(agg[i,f] + b[f])
// ---------------------------------------------------------------------------
template <int F>
__global__ void bias_relu_kernel(const float* __restrict__ agg,
                                 const float* __restrict__ b,
                                 float* __restrict__ out, int n) {
  unsigned idx = blockIdx.x * blockDim.x + threadIdx.x;
  if (idx >= (unsigned)n * F) return;
  unsigned f = idx & (F - 1);
  out[idx] = fmaxf(agg[idx] + b[f], 0.0f);
}

// ---------------------------------------------------------------------------
// Global mean pool + linear head
// ---------------------------------------------------------------------------
__global__ void pool_zero_kernel(float* __restrict__ pool,
                                 float* __restrict__ cnt, int G) {
  int idx = blockIdx.x * blockDim.x + threadIdx.x;
  if (idx < G * 32) pool[idx] = 0.0f;
  if (idx < G) cnt[idx] = 0.0f;
}

__global__ void pool_scatter_kernel(const float* __restrict__ h,
                                    const int* __restrict__ batch,
                                    float* __restrict__ pool,
                                    float* __restrict__ cnt, int n) {
  unsigned idx = blockIdx.x * blockDim.x + threadIdx.x;
  if (idx >= (unsigned)n * 32) return;
  unsigned i = idx >> 5;
  unsigned f = idx & 31;
  int g = batch[i];
  atomicAdd(&pool[(size_t)g * 32 + f], h[idx]);
  if (f == 0) atomicAdd(&cnt[g], 1.0f);
}

__global__ void head_kernel(const float* __restrict__ pool,
                            const float* __restrict__ cnt,
                            const float* __restrict__ Wl,
                            const float* __restrict__ bl,
                            float* __restrict__ out, int G) {
  int g = blockIdx.x * blockDim.x + threadIdx.x;
  if (g >= G) return;
  float inv = 1.0f / fmaxf(cnt[g], 1.0f);
  float a0 = bl[0], a1 = bl[1];
#pragma unroll
  for (int f = 0; f < 32; ++f) {
    float v = pool[g * 32 + f] * inv;
    a0 += v * Wl[f * 2 + 0];
    a1 += v * Wl[f * 2 + 1];
  }
  out[g * 2 + 0] = a0;
  out[g * 2 + 1] = a1;
}

// ---------------------------------------------------------------------------
// Orchestration
// ---------------------------------------------------------------------------
static inline int cdiv(long long a, int b) { return (int)((a + b - 1) / b); }

extern "C" void kernel_launch(void* const* d_in, const int* in_sizes, int n_in,
                              void* d_out, int out_size, void* d_ws, size_t ws_size,
                              hipStream_t stream) {
  const float* x    = (const float*)d_in[0];
  const int*   edge = (const int*)d_in[1];   // JAX x64 disabled -> int32
  const int*   batch= (const int*)d_in[2];
  const float* W1 = (const float*)d_in[3];
  const float* b1 = (const float*)d_in[4];
  const float* W2 = (const float*)d_in[5];
  const float* b2 = (const float*)d_in[6];
  const float* W3 = (const float*)d_in[7];
  const float* b3 = (const float*)d_in[8];
  const float* Wl = (const float*)d_in[9];
  const float* bl = (const float*)d_in[10];
  float* out = (float*)d_out;

  const int N = in_sizes[0] / 5;
  const int E = in_sizes[1] / 2;
  const int G = out_size / 2;
  const int* src = edge;
  const int* dst = edge + E;

  // Workspace layout (floats): dinv | bufA | bufB | pool | cnt
  float* ws = (float*)d_ws;
  size_t nPad = ((size_t)N + 255) & ~(size_t)255;
  float* dinv = ws;
  float* bufA = dinv + nPad;
  float* bufB = bufA + nPad * 64;
  float* pool = bufB + nPad * 64;
  float* cnt  = pool + (size_t)G * 32;

  const int T = 256;
  const int gemmBlocks = cdiv(N / 16, 8);      // one wave per 16-row stripe

  // --- degrees / normalization ---
  deg_init_kernel<<<cdiv(N, T), T, 0, stream>>>(dinv, N);
  deg_count_kernel<<<cdiv(E, T), T, 0, stream>>>(dst, dinv, E);
  deg_finalize_kernel<<<cdiv(N, T), T, 0, stream>>>(dinv, N);

  // --- Layer 1: x(5) -> bufA(64); aggregate in bufB; relu -> bufA ---
  gemm_wmma_kernel<5, 64><<<gemmBlocks, T, 0, stream>>>(x, W1, bufA, N);
  self_init_kernel<64><<<cdiv((long long)N * 64, T), T, 0, stream>>>(bufA, dinv, bufB, N);
  edge_scatter_kernel<64><<<cdiv((long long)E * 16, T), T, 0, stream>>>(bufA, bufB, src, dst, dinv, E);
  bias_relu_kernel<64><<<cdiv((long long)N * 64, T), T, 0, stream>>>(bufB, b1, bufA, N);

  // --- Layer 2: bufA(64) -> bufB(64); aggregate in bufA; relu -> bufB ---
  gemm_wmma_kernel<64, 64><<<gemmBlocks, T, 0, stream>>>(bufA, W2, bufB, N);
  self_init_kernel<64><<<cdiv((long long)N * 64, T), T, 0, stream>>>(bufB, dinv, bufA, N);
  edge_scatter_kernel<64><<<cdiv((long long)E * 16, T), T, 0, stream>>>(bufB, bufA, src, dst, dinv, E);
  bias_relu_kernel<64><<<cdiv((long long)N * 64, T), T, 0, stream>>>(bufA, b2, bufB, N);

  // --- Layer 3: bufB(64) -> bufA(32); aggregate in bufB; relu -> bufA ---
  gemm_wmma_kernel<64, 32><<<gemmBlocks, T, 0, stream>>>(bufB, W3, bufA, N);
  self_init_kernel<32><<<cdiv((long long)N * 32, T), T, 0, stream>>>(bufA, dinv, bufB, N);
  edge_scatter_kernel<32><<<cdiv((long long)E * 8, T), T, 0, stream>>>(bufA, bufB, src, dst, dinv, E);
  bias_relu_kernel<32><<<cdiv((long long)N * 32, T), T, 0, stream>>>(bufB, b3, bufA, N);

  // --- Pool + head ---
  pool_zero_kernel<<<cdiv((long long)G * 32, T), T, 0, stream>>>(pool, cnt, G);
  pool_scatter_kernel<<<cdiv((long long)N * 32, T), T, 0, stream>>>(bufA, batch, pool, cnt, N);
  head_kernel<<<cdiv(G, T), T, 0, stream>>>(pool, cnt, Wl, bl, out, G);
}